// CentroidEuclideanDist_64914135712549
// MI455X (gfx1250) — compile-verified
//
#include <hip/hip_runtime.h>

// dist[b,c] = ||feat[b]||^2 + ||centers[c]||^2 - 2 * <feat[b], centers[c]>
// feat:    [16384, 1024] f32   centers: [4096, 1024] f32   out: [16384, 4096] f32

typedef __attribute__((ext_vector_type(2))) float v2f;
typedef __attribute__((ext_vector_type(8))) float v8f;

#define M_DIM 16384
#define N_DIM 4096
#define D_DIM 1024
#define TK    32                 // K-slab staged through LDS
#define LDSS  36                 // padded LDS row stride (floats)
#define SLAB  (128 * LDSS)       // one 128 x TK tile (padded), in floats
#define NKT   (D_DIM / TK)

#if __has_builtin(__builtin_amdgcn_global_load_async_to_lds_b128)
#define USE_ASYNC_LDS 1
// Exact param type per hipcc diagnostic: int __attribute__((vector_size(16))) AS1*
typedef int v4i __attribute__((vector_size(4 * sizeof(int))));
typedef __attribute__((address_space(1))) v4i g_v4i;   // global (AS1)
typedef __attribute__((address_space(3))) v4i l_v4i;   // LDS (AS3)
#endif

// ---------------------------------------------------------------------------
// Row squared-norm: one wave32 per row, 8 rows per 256-thread block.
// ---------------------------------------------------------------------------
__global__ __launch_bounds__(256) void sqnorm_kernel(const float* __restrict__ x,
                                                     float* __restrict__ out) {
    const int row  = blockIdx.x * 8 + (threadIdx.x >> 5);
    const int lane = threadIdx.x & 31;
    const float4* p = (const float4*)(x + (size_t)row * D_DIM);
    float s = 0.f;
#pragma unroll
    for (int i = 0; i < D_DIM / 4 / 32; ++i) {
        float4 v = p[lane + i * 32];
        s += v.x * v.x + v.y * v.y + v.z * v.z + v.w * v.w;
    }
#pragma unroll
    for (int off = 16; off > 0; off >>= 1) s += __shfl_down(s, off, 32);
    if (lane == 0) out[row] = s;
}

// ---------------------------------------------------------------------------
// Inner MMA over one K-slab: 8 k-steps x (4x2) 16x16 tiles per wave.
// ---------------------------------------------------------------------------
__device__ __forceinline__ void slab_mma(const float* cA, const float* cB,
                                         v8f (&acc)[4][2],
                                         int wm, int wn, int half, int l16) {
#pragma unroll
    for (int ks = 0; ks < TK / 4; ++ks) {
        const int kk = ks * 4 + 2 * half;
        v2f bfr[2];
#pragma unroll
        for (int j = 0; j < 2; ++j)
            bfr[j] = *(const v2f*)&cB[(wn * 32 + j * 16 + l16) * LDSS + kk];
#pragma unroll
        for (int i = 0; i < 4; ++i) {
            v2f afr = *(const v2f*)&cA[(wm * 64 + i * 16 + l16) * LDSS + kk];
#pragma unroll
            for (int j = 0; j < 2; ++j)
                acc[i][j] = __builtin_amdgcn_wmma_f32_16x16x4_f32(
                    false, afr, false, bfr[j], (short)0, acc[i][j], false, false);
        }
    }
}

#ifdef USE_ASYNC_LDS
// Issue one 128xTK A-slab + B-slab as 8 async b128 copies (per thread).
__device__ __forceinline__ void slab_copy_async(const float* ga, const float* gb,
                                                float* dA, float* dB,
                                                int lr, int lc, int kbase) {
#pragma unroll
    for (int p = 0; p < 4; ++p) {
        const float* gsa = ga + (size_t)(p * 32) * D_DIM + kbase;
        const float* gsb = gb + (size_t)(p * 32) * D_DIM + kbase;
        float* lsa = &dA[(lr + p * 32) * LDSS + lc];
        float* lsb = &dB[(lr + p * 32) * LDSS + lc];
        __builtin_amdgcn_global_load_async_to_lds_b128((g_v4i*)gsa, (l_v4i*)lsa, 0, 0);
        __builtin_amdgcn_global_load_async_to_lds_b128((g_v4i*)gsb, (l_v4i*)lsb, 0, 0);
    }
}

__device__ __forceinline__ void wait_async_le8() {
#if __has_builtin(__builtin_amdgcn_s_wait_asynccnt)
    __builtin_amdgcn_s_wait_asynccnt(8);
#else
    asm volatile("s_wait_asynccnt 0x8" ::: "memory");
#endif
}
__device__ __forceinline__ void wait_async_le0() {
#if __has_builtin(__builtin_amdgcn_s_wait_asynccnt)
    __builtin_amdgcn_s_wait_asynccnt(0);
#else
    asm volatile("s_wait_asynccnt 0x0" ::: "memory");
#endif
}
#endif

// ---------------------------------------------------------------------------
// Fused distance GEMM: 128x128 block tile, 8 waves (2x4), wave tile 64x32,
// f32 WMMA 16x16x4, async double-buffered LDS staging, fused norm epilogue.
// ---------------------------------------------------------------------------
__global__ __launch_bounds__(256) void centroid_dist_kernel(
        const float* __restrict__ A,    // feat    [M, D]
        const float* __restrict__ Bm,   // centers [N, D]
        const float* __restrict__ x2,   // [M]
        const float* __restrict__ c2,   // [N]
        float* __restrict__ out) {      // [M, N]

#ifdef USE_ASYNC_LDS
    __shared__ float sA[2 * SLAB];
    __shared__ float sB[2 * SLAB];
#else
    __shared__ float sA[SLAB];
    __shared__ float sB[SLAB];
#endif

    const int t  = threadIdx.x;
    const int bn = blockIdx.x;    // N / 128 = 32
    const int bm = blockIdx.y;    // M / 128 = 128

    // Global->LDS mapping: thread t handles float4 at (row = t>>3 + 32p, col = (t&7)*4)
    const int lr = t >> 3;
    const int lc = (t & 7) << 2;
    const float* ga = A  + (size_t)(bm * 128 + lr) * D_DIM + lc;
    const float* gb = Bm + (size_t)(bn * 128 + lr) * D_DIM + lc;

    // Wave/lane mapping (wave32): 8 waves as 2 (M) x 4 (N)
    const int wave = t >> 5;
    const int lane = t & 31;
    const int wm   = wave >> 2;
    const int wn   = wave & 3;
    const int half = lane >> 4;
    const int l16  = lane & 15;

    v8f acc[4][2] = {};

#ifdef USE_ASYNC_LDS
    // Double-buffered pipeline on ASYNCcnt: slab k+1 copies overlap slab k MMA.
    slab_copy_async(ga, gb, sA, sB, lr, lc, 0);
    for (int kt = 0; kt < NKT; ++kt) {
        const int cur = kt & 1;
        if (kt + 1 < NKT) {
            slab_copy_async(ga, gb, sA + (cur ^ 1) * SLAB, sB + (cur ^ 1) * SLAB,
                            lr, lc, (kt + 1) * TK);
            wait_async_le8();      // async loads retire in order: slab kt has landed
        } else {
            wait_async_le0();
        }
        __syncthreads();
        slab_mma(sA + cur * SLAB, sB + cur * SLAB, acc, wm, wn, half, l16);
        __syncthreads();           // everyone done reading before buffer is rewritten
    }
#else
    for (int kt = 0; kt < NKT; ++kt) {
        const int kbase = kt * TK;
#pragma unroll
        for (int p = 0; p < 4; ++p) {
            float4 va = *(const float4*)(ga + (size_t)(p * 32) * D_DIM + kbase);
            float4 vb = *(const float4*)(gb + (size_t)(p * 32) * D_DIM + kbase);
            *(float4*)&sA[(lr + p * 32) * LDSS + lc] = va;
            *(float4*)&sB[(lr + p * 32) * LDSS + lc] = vb;
        }
        if (kt + 1 < NKT) {
            __builtin_prefetch(ga + kbase + TK, 0, 3);
            __builtin_prefetch(gb + kbase + TK, 0, 3);
        }
        __syncthreads();
        slab_mma(sA, sB, acc, wm, wn, half, l16);
        __syncthreads();
    }
#endif

    // Epilogue: dist = x2[m] + c2[n] - 2*dot   (C/D layout: m = v + 8*half, n = l16)
    const int nbase = bn * 128 + wn * 32;
    const int mbase = bm * 128 + wm * 64;
#pragma unroll
    for (int i = 0; i < 4; ++i) {
#pragma unroll
        for (int j = 0; j < 2; ++j) {
            const int n  = nbase + j * 16 + l16;
            const float cn = c2[n];
#pragma unroll
            for (int v = 0; v < 8; ++v) {
                const int m = mbase + i * 16 + half * 8 + v;
                out[(size_t)m * N_DIM + n] = x2[m] + cn - 2.0f * acc[i][j][v];
            }
        }
    }
}

// ---------------------------------------------------------------------------
extern "C" void kernel_launch(void* const* d_in, const int* in_sizes, int n_in,
                              void* d_out, int out_size, void* d_ws, size_t ws_size,
                              hipStream_t stream) {
    (void)in_sizes; (void)n_in; (void)out_size; (void)ws_size;
    const float* feat    = (const float*)d_in[0];
    const float* centers = (const float*)d_in[1];
    float* out = (float*)d_out;

    float* x2 = (float*)d_ws;          // [16384]
    float* c2 = x2 + M_DIM;            // [4096]

    sqnorm_kernel<<<M_DIM / 8, 256, 0, stream>>>(feat, x2);
    sqnorm_kernel<<<N_DIM / 8, 256, 0, stream>>>(centers, c2);

    dim3 grid(N_DIM / 128, M_DIM / 128);
    centroid_dist_kernel<<<grid, 256, 0, stream>>>(feat, centers, x2, c2, out);
}